// PRQTransform_84473416777847
// MI455X (gfx1250) — compile-verified
//
#include <hip/hip_runtime.h>
#include <hip/hip_bf16.h>
#include <stdint.h>

#define NUM_BINS 10
#define THREADS  256

typedef uint32_t u32x4 __attribute__((ext_vector_type(4)));
typedef int32_t  i32x8 __attribute__((ext_vector_type(8)));
typedef int32_t  i32x4 __attribute__((ext_vector_type(4)));

// log(exp(0.999) - 1): the constant the reference writes into ud[0], ud[10]
#define DCONST 0.53974243f

// Issue one 1-D TDM DMA: n_elems f32 from gptr -> LDS byte offset lds_byte_off.
// D# layout per CDNA5 ISA ch.8 (group0 128b, group1 256b; groups 2/3 zero).
// This toolchain exposes the 6-arg builtin: (g0, g1, g2, g3, g_extra, cpol).
__device__ __forceinline__ void tdm_load_1d_f32(uint32_t lds_byte_off,
                                                const float* gptr,
                                                uint32_t n_elems) {
  uint64_t ga = (uint64_t)(uintptr_t)gptr;
  u32x4 g0;
  g0[0] = 1u;                                                // count=1 (valid), user mode
  g0[1] = lds_byte_off;                                      // lds_addr (bytes)
  g0[2] = (uint32_t)ga;                                      // global_addr[31:0]
  g0[3] = (uint32_t)((ga >> 32) & 0x01FFFFFFu) | (2u << 30); // global_addr[56:32] | type=2
  i32x8 g1;
  g1[0] = (int32_t)(2u << 16);                               // data_size=2 -> 4 bytes
  g1[1] = (int32_t)((n_elems & 0xFFFFu) << 16);              // tensor_dim0[15:0]
  g1[2] = (int32_t)(((n_elems >> 16) & 0xFFFFu) | (1u << 16)); // tensor_dim0[31:16], tensor_dim1=1
  g1[3] = (int32_t)((n_elems & 0xFFFFu) << 16);              // tile_dim0 = n_elems (<= 2816, fits)
  g1[4] = 0;                                                 // tile_dim1=0, tile_dim2=0 (unused)
  g1[5] = (int32_t)n_elems;                                  // tensor_dim0_stride[31:0]
  g1[6] = 0;                                                 // stride hi, dim1_stride lo
  g1[7] = 0;
  i32x4 z4 = {0, 0, 0, 0};
  i32x8 z8 = {0, 0, 0, 0, 0, 0, 0, 0};
  __builtin_amdgcn_tensor_load_to_lds(g0, g1, z4, z4, z8, 0);
}

__global__ __launch_bounds__(THREADS) void
prqs_inverse_kernel(const float* __restrict__ x_in,
                    const float* __restrict__ uw_in,
                    const float* __restrict__ uh_in,
                    const float* __restrict__ ud_in,
                    float* __restrict__ out, int n) {
  __shared__ float s_uw[THREADS * NUM_BINS];        // 10240 B
  __shared__ float s_uh[THREADS * NUM_BINS];        // 10240 B
  __shared__ float s_ud[THREADS * (NUM_BINS + 1)];  // 11264 B

  const int base  = blockIdx.x * THREADS;
  const int count = min(THREADS, n - base);

  // Wave 0 issues three TDM DMAs covering this block's tiles, waits TENSORcnt.
  if (threadIdx.x < 32) {
    tdm_load_1d_f32((uint32_t)(uintptr_t)s_uw,
                    uw_in + (size_t)base * NUM_BINS,
                    (uint32_t)(count * NUM_BINS));
    tdm_load_1d_f32((uint32_t)(uintptr_t)s_uh,
                    uh_in + (size_t)base * NUM_BINS,
                    (uint32_t)(count * NUM_BINS));
    tdm_load_1d_f32((uint32_t)(uintptr_t)s_ud,
                    ud_in + (size_t)base * (NUM_BINS + 1),
                    (uint32_t)(count * (NUM_BINS + 1)));
    __builtin_amdgcn_s_wait_tensorcnt(0);
  }
  __syncthreads();

  const int t = threadIdx.x;
  if (t >= count) return;
  const int gi = base + t;
  const float x = x_in[gi];

  const float* uwp = &s_uw[t * NUM_BINS];
  const float* uhp = &s_uh[t * NUM_BINS];
  const float* udp = &s_ud[t * (NUM_BINS + 1)];

  // --- softmax (max-subtracted) for widths & heights ---
  float mw = uwp[0], mh = uhp[0];
#pragma unroll
  for (int k = 1; k < NUM_BINS; ++k) {
    mw = fmaxf(mw, uwp[k]);
    mh = fmaxf(mh, uhp[k]);
  }
  float ew[NUM_BINS], eh[NUM_BINS];
  float sw = 0.f, sh = 0.f;
#pragma unroll
  for (int k = 0; k < NUM_BINS; ++k) {
    ew[k] = __expf(uwp[k] - mw); sw += ew[k];
    eh[k] = __expf(uhp[k] - mh); sh += eh[k];
  }
  const float rsw = 1.0f / sw;
  const float rsh = 1.0f / sh;

  // --- fused cumsum + knot positions + bin select (last k with x >= ch_k) ---
  float cumw = 0.f, cumh = 0.f;
  float cwk = -5.f, chk = -5.f;
  float ich = -5.f, ichn = 5.f, icw = -5.f, icwn = 5.f;
  float iud0 = DCONST, iud1 = DCONST;
#pragma unroll
  for (int k = 0; k < NUM_BINS; ++k) {
    cumw += fmaf(0.99f, ew[k] * rsw, 1e-3f);   // width fraction, sums ~1
    cumh += fmaf(0.99f, eh[k] * rsh, 1e-3f);   // height fraction
    const float cwn = (k == NUM_BINS - 1) ? 5.f : fmaf(10.f, cumw, -5.f);
    const float chn = (k == NUM_BINS - 1) ? 5.f : fmaf(10.f, cumh, -5.f);
    const float u0 = (k == 0)            ? DCONST : udp[k];
    const float u1 = (k == NUM_BINS - 1) ? DCONST : udp[k + 1];
    const bool c = (x >= chk);                 // k==0: always true (chk=-5, x>-4.9)
    ich  = c ? chk : ich;   ichn = c ? chn : ichn;
    icw  = c ? cwk : icw;   icwn = c ? cwn : icwn;
    iud0 = c ? u0  : iud0;  iud1 = c ? u1  : iud1;
    cwk = cwn; chk = chn;
  }

  // --- inverse rational-quadratic solve ---
  const float iw = icwn - icw;
  const float ih = ichn - ich;
  const float idelta = ih / iw;
  // softplus (stable): max(v,0) + log1p(exp(-|v|)); only 2 evaluations needed
  const float d0 = 1e-3f + fmaxf(iud0, 0.f) + log1pf(__expf(-fabsf(iud0)));
  const float d1 = 1e-3f + fmaxf(iud1, 0.f) + log1pf(__expf(-fabsf(iud1)));

  const float dy = x - ich;
  const float s2 = d0 + d1 - 2.f * idelta;
  const float aq = dy * s2 + ih * (idelta - d0);
  const float bq = ih * d0 - dy * s2;
  const float cq = -idelta * dy;
  const float disc = bq * bq - 4.f * aq * cq;
  const float root = (2.f * cq) / (-bq - sqrtf(disc));
  out[gi] = fmaf(root, iw, icw);
}

extern "C" void kernel_launch(void* const* d_in, const int* in_sizes, int n_in,
                              void* d_out, int out_size, void* d_ws, size_t ws_size,
                              hipStream_t stream) {
  const float* x  = (const float*)d_in[0];
  const float* uw = (const float*)d_in[1];
  const float* uh = (const float*)d_in[2];
  const float* ud = (const float*)d_in[3];
  float* out = (float*)d_out;
  const int n = in_sizes[0];
  const int blocks = (n + THREADS - 1) / THREADS;
  prqs_inverse_kernel<<<blocks, THREADS, 0, stream>>>(x, uw, uh, ud, out, n);
}